// ArrayNDSubdivided_88536455839953
// MI455X (gfx1250) — compile-verified
//
#include <hip/hip_runtime.h>
#include <cstdint>

// Problem constants (from the reference)
#define NPTS        4000000
#define TW          128            // tensor W
#define THH         8192           // tensor H (= 4*4*4 voxels * 128)
#define NCH         32
#define NTEX        (TW * THH)     // texels per channel plane (1,048,576)
#define CH_STRIDE_B (NTEX * 4)     // 4 MiB per channel plane (CHW layout)
#define TPB         256

typedef float    f32x4    __attribute__((ext_vector_type(4)));
typedef unsigned uint32x4 __attribute__((ext_vector_type(4)));
typedef int      int32x4  __attribute__((ext_vector_type(4)));
typedef int      int32x8  __attribute__((ext_vector_type(8)));

// unaligned (4B) pair load for the CHW fallback path
struct __attribute__((packed, aligned(4))) F2 { float lo, hi; };

#if __has_builtin(__builtin_amdgcn_tensor_load_to_lds) && __has_builtin(__builtin_amdgcn_s_wait_tensorcnt)
#define HAVE_TDM 1
#else
#define HAVE_TDM 0
#endif

static __device__ __forceinline__ int iclamp(int v, int lo, int hi) {
    return v < lo ? lo : (v > hi ? hi : v);
}

// -------------------------------------------------------------------------
// Shared per-point setup: TDM coordinate staging + bilinear weights/indices
// -------------------------------------------------------------------------
struct PointSetup {
    float w00, w01, w10, w11;   // validity & mask folded in
    int   ix0, iy0;             // raw floor indices (pre-clamp)
};

static __device__ __forceinline__ void stage_coords_tdm(const float* __restrict__ x,
                                                        float* sx /* LDS, TPB*3 */)
{
#if HAVE_TDM
    if ((threadIdx.x >> 5) == 0) {   // wave 0 issues exactly one tensor_load_to_lds
        const uint64_t ga  = (uint64_t)(uintptr_t)(x + (size_t)blockIdx.x * (TPB * 3));
        const uint32_t lds = (uint32_t)(uintptr_t)(void*)sx;

        uint32x4 g0;                                            // D# group 0
        g0[0] = 1u;                                             // count=1, user mode
        g0[1] = lds;                                            // lds_addr
        g0[2] = (uint32_t)(ga & 0xffffffffu);                   // global_addr[31:0]
        g0[3] = (uint32_t)((ga >> 32) & 0x1ffffffu) | (2u << 30); // [56:32], type=2

        const uint32_t td0 = (uint32_t)(TPB * 3);
        int32x8 g1;                                             // D# group 1
        g1[0] = 0x20000;                                        // data_size=2 (4B)
        g1[1] = (int)((td0 & 0xffffu) << 16);                   // tensor_dim0[15:0]
        g1[2] = (int)((td0 >> 16) & 0xffffu) | (int)(1u << 16); // tensor_dim0[31:16], tensor_dim1=1
        g1[3] = (int)(td0 << 16);                               // tile_dim0 = 768
        g1[4] = 1;                                              // tile_dim1 = 1
        g1[5] = (int)td0;                                       // tensor_dim0_stride
        g1[6] = 0;
        g1[7] = 0;

        int32x4 gz = {0, 0, 0, 0};
#if defined(__clang_major__) && (__clang_major__ >= 23)
        int32x8 gz8 = {0, 0, 0, 0, 0, 0, 0, 0};
        __builtin_amdgcn_tensor_load_to_lds(g0, g1, gz, gz, gz8, 0);
#else
        __builtin_amdgcn_tensor_load_to_lds(g0, g1, gz, gz, 0);
#endif
        __builtin_amdgcn_s_wait_tensorcnt(0);
    }
    __syncthreads();
#else
    (void)x; (void)sx;
#endif
}

static __device__ __forceinline__ PointSetup point_setup(float px, float py, float pz)
{
    // voxel index (reference: clip -> floor(v*4) -> clip 0..3)
    const float cxv = fminf(fmaxf(px, 0.f), 1.f);
    const float cyv = fminf(fmaxf(py, 0.f), 1.f);
    const float czv = fminf(fmaxf(pz, 0.f), 1.f);
    const float vxf = fminf(floorf(cxv * 4.f), 3.f);
    const float vyf = fminf(floorf(cyv * 4.f), 3.f);
    const float vzf = fminf(floorf(czv * 4.f), 3.f);
    const float voxel = vzf * 16.f + vyf * 4.f + vxf;

    const bool inb = (px >= 0.f) && (px <= 1.f) && (py >= 0.f) && (py <= 1.f);
    const float m = inb ? 1.f : 0.f;

    // grid coords, same op order as the reference for bit-closeness
    const float xm0 = fmodf(px * 512.f, 128.f);
    const float xm1 = fmodf(py * 512.f, 128.f);
    const float gx  = xm0 / 128.f * 2.f - 1.f;
    const float gy  = (xm1 + voxel * 128.f) / 8192.f * 2.f - 1.f;
    const float ixf = ((gx + 1.f) * 128.f  - 1.f) * 0.5f;
    const float iyf = ((gy + 1.f) * 8192.f - 1.f) * 0.5f;

    const float fx0 = floorf(ixf);
    const float fy0 = floorf(iyf);
    const float wx  = ixf - fx0;
    const float wy  = iyf - fy0;

    PointSetup s;
    s.ix0 = (int)fx0;
    s.iy0 = (int)fy0;

    const float vx0 = (s.ix0 >= 0  && s.ix0 <= TW - 1)  ? 1.f : 0.f;
    const float vx1 = (s.ix0 >= -1 && s.ix0 <= TW - 2)  ? 1.f : 0.f;
    const float vy0 = (s.iy0 >= 0  && s.iy0 <= THH - 1) ? 1.f : 0.f;
    const float vy1 = (s.iy0 >= -1 && s.iy0 <= THH - 2) ? 1.f : 0.f;

    s.w00 = (1.f - wx) * (1.f - wy) * (vx0 * vy0 * m);
    s.w01 = wx         * (1.f - wy) * (vx1 * vy0 * m);
    s.w10 = (1.f - wx) * wy         * (vx0 * vy1 * m);
    s.w11 = wx         * wy         * (vx1 * vy1 * m);
    return s;
}

// -------------------------------------------------------------------------
// Pass 0 (fast path): CHW -> HWC transpose into scratch.
// Reads non-temporal (stream once, keep out of L2); writes regular-temporal
// so the transposed copy becomes the L2-resident gather target.
// -------------------------------------------------------------------------
__global__ __launch_bounds__(TPB) void
transpose_chw_hwc(const float* __restrict__ src, float* __restrict__ dst)
{
    const uint32_t i = blockIdx.x * TPB + threadIdx.x;  // texel index
    float v[NCH];
#pragma unroll
    for (int c = 0; c < NCH; ++c)
        v[c] = __builtin_nontemporal_load(src + (size_t)c * NTEX + i);

    float* d = dst + (size_t)i * NCH;
#pragma unroll
    for (int k = 0; k < NCH / 4; ++k) {
        f32x4 r = { v[4 * k], v[4 * k + 1], v[4 * k + 2], v[4 * k + 3] };
        *(f32x4*)(d + 4 * k) = r;   // RT store: want this resident in L2
    }
}

// -------------------------------------------------------------------------
// Fast sampler: HWC layout. One corner = one contiguous, 16B-aligned 128B
// channel vector -> 32x global_load_b128 + 8x NT global_store_b128 per point.
// -------------------------------------------------------------------------
__global__ __launch_bounds__(TPB) void
sample_hwc(const float* __restrict__ x,
           const float* __restrict__ tt,    // (H, W, C)
           float* __restrict__ out)
{
    const int tid = threadIdx.x;
    const int n   = blockIdx.x * TPB + tid;

#if HAVE_TDM
    __shared__ float sx[TPB * 3];
    stage_coords_tdm(x, sx);
    const float px = sx[tid * 3 + 0];
    const float py = sx[tid * 3 + 1];
    const float pz = sx[tid * 3 + 2];
#else
    const float px = x[(size_t)n * 3 + 0];
    const float py = x[(size_t)n * 3 + 1];
    const float pz = x[(size_t)n * 3 + 2];
#endif
    if (n >= NPTS) return;

    const PointSetup s = point_setup(px, py, pz);

    const int x0 = iclamp(s.ix0,     0, TW - 1);
    const int x1 = iclamp(s.ix0 + 1, 0, TW - 1);
    const int y0 = iclamp(s.iy0,     0, THH - 1);
    const int y1 = iclamp(s.iy0 + 1, 0, THH - 1);

    const f32x4* p00 = (const f32x4*)(tt + (size_t)(y0 * TW + x0) * NCH);
    const f32x4* p01 = (const f32x4*)(tt + (size_t)(y0 * TW + x1) * NCH);
    const f32x4* p10 = (const f32x4*)(tt + (size_t)(y1 * TW + x0) * NCH);
    const f32x4* p11 = (const f32x4*)(tt + (size_t)(y1 * TW + x1) * NCH);

    float* op = out + (size_t)n * NCH;
#pragma unroll
    for (int k = 0; k < NCH / 4; ++k) {
        f32x4 r = p00[k] * s.w00 + p01[k] * s.w01 + p10[k] * s.w10 + p11[k] * s.w11;
        __builtin_nontemporal_store(r, (f32x4*)(op + k * 4));
    }
}

// -------------------------------------------------------------------------
// Fallback sampler (scratch too small): CHW layout with b64 pair loads and
// once-per-point edge weight remap (round-1 kernel, verified codegen).
// -------------------------------------------------------------------------
__global__ __launch_bounds__(TPB) void
sample_chw(const float* __restrict__ x,
           const float* __restrict__ tensor,
           float* __restrict__ out)
{
    const int tid = threadIdx.x;
    const int n   = blockIdx.x * TPB + tid;

#if HAVE_TDM
    __shared__ float sx[TPB * 3];
    stage_coords_tdm(x, sx);
    const float px = sx[tid * 3 + 0];
    const float py = sx[tid * 3 + 1];
    const float pz = sx[tid * 3 + 2];
#else
    const float px = x[(size_t)n * 3 + 0];
    const float py = x[(size_t)n * 3 + 1];
    const float pz = x[(size_t)n * 3 + 2];
#endif
    if (n >= NPTS) return;

    const PointSetup s = point_setup(px, py, pz);

    float wlt = s.w00, wht = s.w01, wlb = s.w10, whb = s.w11;
    if (s.ix0 < 0)           { wlt = s.w01; wht = 0.f; wlb = s.w11; whb = 0.f; }
    else if (s.ix0 > TW - 2) { wht = s.w00; wlt = 0.f; whb = s.w10; wlb = 0.f; }

    const int xb = iclamp(s.ix0, 0, TW - 2);
    const int yt = iclamp(s.iy0,     0, THH - 1);
    const int yb = iclamp(s.iy0 + 1, 0, THH - 1);
    const uint32_t ot = (uint32_t)(yt * TW + xb) * 4u;
    const uint32_t ob = (uint32_t)(yb * TW + xb) * 4u;

    const char* tb = (const char*)tensor;
    float* op = out + (size_t)n * NCH;

#pragma unroll
    for (int c4 = 0; c4 < NCH / 4; ++c4) {
        float rv[4];
#pragma unroll
        for (int j = 0; j < 4; ++j) {
            const uint32_t coff = (uint32_t)(c4 * 4 + j) * (uint32_t)CH_STRIDE_B;
            const F2 t = *(const F2*)(tb + (coff + ot));
            const F2 b = *(const F2*)(tb + (coff + ob));
            rv[j] = wlt * t.lo + wht * t.hi + wlb * b.lo + whb * b.hi;
        }
        f32x4 r = { rv[0], rv[1], rv[2], rv[3] };
        __builtin_nontemporal_store(r, (f32x4*)(op + c4 * 4));
    }
}

// -------------------------------------------------------------------------
extern "C" void kernel_launch(void* const* d_in, const int* in_sizes, int n_in,
                              void* d_out, int out_size, void* d_ws, size_t ws_size,
                              hipStream_t stream) {
    (void)in_sizes; (void)n_in; (void)out_size;
    const float* x      = (const float*)d_in[0];
    const float* tensor = (const float*)d_in[1];
    float* out          = (float*)d_out;

    const int sample_blocks = (NPTS + TPB - 1) / TPB;       // 15625
    const size_t needed = (size_t)NCH * NTEX * sizeof(float); // 128 MiB

    if (ws_size >= needed && d_ws != nullptr) {
        float* tt = (float*)d_ws;
        const int tr_blocks = NTEX / TPB;                   // 4096
        transpose_chw_hwc<<<dim3(tr_blocks), dim3(TPB), 0, stream>>>(tensor, tt);
        sample_hwc<<<dim3(sample_blocks), dim3(TPB), 0, stream>>>(x, tt, out);
    } else {
        sample_chw<<<dim3(sample_blocks), dim3(TPB), 0, stream>>>(x, tensor, out);
    }
}